// MultiTaskGAT_1958505087789
// MI455X (gfx1250) — compile-verified
//
#include <hip/hip_runtime.h>

// ---------------- problem constants ----------------
constexpr int NN  = 20000;    // nodes
constexpr int NE  = 320000;   // edges
constexpr int NG  = 64;       // graphs
constexpr int H1  = 4;        // heads layer 1
constexpr int HID = 128;
constexpr float NEG_SLOPE = 0.2f;

typedef __attribute__((ext_vector_type(2))) float v2f;
typedef __attribute__((ext_vector_type(8))) float v8f;

// ---------------- utility kernels ----------------
__global__ void fill_f32(float* __restrict__ p, float v, size_t n) {
    size_t i = (size_t)blockIdx.x * blockDim.x + threadIdx.x;
    if (i < n) p[i] = v;
}

__device__ __forceinline__ void atomicMaxF(float* addr, float val) {
    // IEEE-754 ordered-compare trick; works with -inf init and mixed signs.
    if (val >= 0.0f) atomicMax((int*)addr, __float_as_int(val));
    else             atomicMin((unsigned int*)addr, __float_as_uint(val));
}

// ---------------- WMMA fp32 GEMM core ----------------
// wave tile = 32(M) x 64(N): acc[2][4] v8f, 8 WMMAs per K-step of 4.
// block = 256 threads = 8 waves as 4(M) x 2(N) -> block tile 128 x 128.
// grid = (ceil(M/128), N/128)

#define WMMA_F32(a, b, c) \
    __builtin_amdgcn_wmma_f32_16x16x4_f32(false, (a), false, (b), (short)0, (c), false, false)

// Straight-line K-segment loop. A0/A1 already point at (row, segment_start + 2*half).
// Bp0 = B + (size_t)(kBeg + 2*half) * N + colBase + l16.
__device__ __forceinline__ void gemm_seg(const float* __restrict__ A0,
                                         const float* __restrict__ A1,
                                         const float* __restrict__ Bp0,
                                         int N, int kLen, v8f acc[2][4]) {
    for (int kk = 0; kk < kLen; kk += 4) {
        v2f a0 = *(const v2f*)(A0 + kk);
        v2f a1 = *(const v2f*)(A1 + kk);
        const float* Bp = Bp0 + (size_t)kk * N;
        v2f b[4];
#pragma unroll
        for (int j = 0; j < 4; ++j) {
            b[j].x = Bp[16 * j];
            b[j].y = Bp[N + 16 * j];
        }
#pragma unroll
        for (int j = 0; j < 4; ++j) {
            acc[0][j] = WMMA_F32(a0, b[j], acc[0][j]);
            acc[1][j] = WMMA_F32(a1, b[j], acc[1][j]);
        }
    }
}

__device__ __forceinline__ void wmma_store_tile(float* __restrict__ C, int ldc, int M,
                                                int row0, int col, const v8f& d,
                                                int half, int l16) {
#pragma unroll
    for (int r = 0; r < 8; ++r) {
        int row = row0 + 8 * half + r;
        if (row < M) C[(size_t)row * ldc + col + l16] = d[r];
    }
}

__device__ __forceinline__ void wmma_store_all(float* __restrict__ C, int N, int M,
                                               int rowBase, int colBase,
                                               v8f acc[2][4], int half, int l16) {
#pragma unroll
    for (int i = 0; i < 2; ++i)
#pragma unroll
        for (int j = 0; j < 4; ++j)
            wmma_store_tile(C, N, M, rowBase + 16 * i, colBase + 16 * j, acc[i][j], half, l16);
}

// GEMM1: C[20000,512] = concat(rf[.,100] | fe[.,768] | cb[.,768]) @ B[1636,512]
__global__ __launch_bounds__(256) void gat_gemm1_wmma(
    const float* __restrict__ rf, const float* __restrict__ fe,
    const float* __restrict__ cb, const float* __restrict__ B,
    float* __restrict__ C) {
    constexpr int M = NN, N = H1 * HID;   // 20000 x 512
    const int wave = threadIdx.x >> 5, lane = threadIdx.x & 31;
    const int wm = wave >> 1, wn = wave & 1;
    const int rowBase = blockIdx.x * 128 + wm * 32;
    const int colBase = blockIdx.y * 128 + wn * 64;
    const int half = lane >> 4, l16 = lane & 15;

    const int ar0 = min(rowBase + l16, M - 1);
    const int ar1 = min(rowBase + 16 + l16, M - 1);

    v8f acc[2][4] = {};
    const float* Bbase = B + colBase + l16 + (size_t)(2 * half) * N;

    // segment 1: randfeat, K = [0, 100)
    gemm_seg(rf + (size_t)ar0 * 100 + 2 * half,
             rf + (size_t)ar1 * 100 + 2 * half,
             Bbase, N, 100, acc);
    // segment 2: func_emb, K = [100, 868)
    gemm_seg(fe + (size_t)ar0 * 768 + 2 * half,
             fe + (size_t)ar1 * 768 + 2 * half,
             Bbase + (size_t)100 * N, N, 768, acc);
    // segment 3: codebert, K = [868, 1636)
    gemm_seg(cb + (size_t)ar0 * 768 + 2 * half,
             cb + (size_t)ar1 * 768 + 2 * half,
             Bbase + (size_t)868 * N, N, 768, acc);

    wmma_store_all(C, N, M, rowBase, colBase, acc, half, l16);
}

// Generic fp32 WMMA GEMM: C[M,N] = A[M,K] @ B[K,N]; K % 4 == 0, N % 128 == 0
__global__ __launch_bounds__(256) void gemm_f32_wmma(
    const float* __restrict__ A, const float* __restrict__ B,
    float* __restrict__ C, int M, int N, int K) {
    const int wave = threadIdx.x >> 5, lane = threadIdx.x & 31;
    const int wm = wave >> 1, wn = wave & 1;
    const int rowBase = blockIdx.x * 128 + wm * 32;
    const int colBase = blockIdx.y * 128 + wn * 64;
    const int half = lane >> 4, l16 = lane & 15;

    const int ar0 = min(rowBase + l16, M - 1);
    const int ar1 = min(rowBase + 16 + l16, M - 1);

    v8f acc[2][4] = {};
    gemm_seg(A + (size_t)ar0 * K + 2 * half,
             A + (size_t)ar1 * K + 2 * half,
             B + colBase + l16 + (size_t)(2 * half) * N,
             N, K, acc);

    wmma_store_all(C, N, M, rowBase, colBase, acc, half, l16);
}

// ---------------- GAT attention kernels ----------------
// el[i]/er[i] for task i = n*H + h : dot(feat[i,:], attn_{l,r}[h,:])
__global__ void node_attn(const float* __restrict__ feat,
                          const float* __restrict__ attn_l,
                          const float* __restrict__ attn_r,
                          float* __restrict__ el, float* __restrict__ er,
                          int n_tasks, int H, int D) {
    int i = blockIdx.x * blockDim.x + threadIdx.x;
    if (i >= n_tasks) return;
    int h = i % H;
    const float* f = feat + (size_t)i * D;
    float sl = 0.f, sr = 0.f;
    for (int d = 0; d < D; ++d) {
        float v = f[d];
        sl += v * attn_l[h * D + d];
        sr += v * attn_r[h * D + d];
    }
    el[i] = sl;
    er[i] = sr;
}

__device__ __forceinline__ float edge_score(const float* el, const float* er,
                                            int s, int d, int h, int H) {
    float x = el[s * H + h] + er[d * H + h];
    return x >= 0.f ? x : NEG_SLOPE * x;
}

__global__ void edge_max_k(const int* __restrict__ src, const int* __restrict__ dst,
                           const float* __restrict__ el, const float* __restrict__ er,
                           float* __restrict__ m, int E, int H) {
    int i = blockIdx.x * blockDim.x + threadIdx.x;
    if (i >= E * H) return;
    int e = i / H, h = i - e * H;
    int s = src[e], d = dst[e];
    atomicMaxF(&m[d * H + h], edge_score(el, er, s, d, h, H));
}

__global__ void edge_sumexp_k(const int* __restrict__ src, const int* __restrict__ dst,
                              const float* __restrict__ el, const float* __restrict__ er,
                              const float* __restrict__ m, float* __restrict__ z,
                              int E, int H) {
    int i = blockIdx.x * blockDim.x + threadIdx.x;
    if (i >= E * H) return;
    int e = i / H, h = i - e * H;
    int s = src[e], d = dst[e];
    float x = edge_score(el, er, s, d, h, H);
    atomicAdd(&z[d * H + h], __expf(x - m[d * H + h]));
}

// one wave per (edge, head); lane-strided scatter-add of feat[src]*alpha into rst[dst]
__global__ __launch_bounds__(256) void edge_aggregate_k(
    const int* __restrict__ src, const int* __restrict__ dst,
    const float* __restrict__ el, const float* __restrict__ er,
    const float* __restrict__ m, const float* __restrict__ z,
    const float* __restrict__ feat, float* __restrict__ rst,
    int E, int H, int D) {
    int gw = (int)(((size_t)blockIdx.x * blockDim.x + threadIdx.x) >> 5);
    int lane = threadIdx.x & 31;
    if (gw >= E * H) return;
    int e = gw / H, h = gw - e * H;
    int s = src[e], d = dst[e];
    float x = edge_score(el, er, s, d, h, H);
    float a = __expf(x - m[d * H + h]) / z[d * H + h];
    const float* fs = feat + ((size_t)s * H + h) * D;
    float* rd = rst + ((size_t)d * H + h) * D;
    for (int c = lane; c < D; c += 32) atomicAdd(&rd[c], fs[c] * a);
}

// x[n, F] += bias[F]; optional ELU
__global__ void bias_act_k(float* __restrict__ x, const float* __restrict__ b,
                           size_t total, int F, int do_elu) {
    size_t i = (size_t)blockIdx.x * blockDim.x + threadIdx.x;
    if (i >= total) return;
    float v = x[i] + b[i % F];
    if (do_elu) v = v > 0.f ? v : __expf(v) - 1.f;
    x[i] = v;
}

// ---------------- classifier heads ----------------
// wave per node: logits = relu(h @ W1[128,64] + b1) @ W2[64,2] + b2
__global__ __launch_bounds__(256) void node_mlp_k(
    const float* __restrict__ h, const float* __restrict__ W1,
    const float* __restrict__ b1, const float* __restrict__ W2,
    const float* __restrict__ b2, float* __restrict__ out, int n) {
    int gw = (int)(((size_t)blockIdx.x * blockDim.x + threadIdx.x) >> 5);
    int lane = threadIdx.x & 31;
    if (gw >= n) return;
    const float* hn = h + (size_t)gw * 128;
    float h0 = b1[2 * lane], h1 = b1[2 * lane + 1];
    for (int k = 0; k < 128; ++k) {
        float v = hn[k];
        h0 += v * W1[k * 64 + 2 * lane];
        h1 += v * W1[k * 64 + 2 * lane + 1];
    }
    h0 = fmaxf(h0, 0.f);
    h1 = fmaxf(h1, 0.f);
    float p0 = h0 * W2[(2 * lane) * 2 + 0] + h1 * W2[(2 * lane + 1) * 2 + 0];
    float p1 = h0 * W2[(2 * lane) * 2 + 1] + h1 * W2[(2 * lane + 1) * 2 + 1];
    for (int off = 16; off > 0; off >>= 1) {
        p0 += __shfl_down(p0, off, 32);
        p1 += __shfl_down(p1, off, 32);
    }
    if (lane == 0) {
        out[gw * 2 + 0] = p0 + b2[0];
        out[gw * 2 + 1] = p1 + b2[1];
    }
}

// wave per node: scatter-add node features + counts per graph
__global__ __launch_bounds__(256) void graph_accum_k(
    const float* __restrict__ h, const int* __restrict__ gid,
    float* __restrict__ gsum, float* __restrict__ cnt, int n) {
    int gw = (int)(((size_t)blockIdx.x * blockDim.x + threadIdx.x) >> 5);
    int lane = threadIdx.x & 31;
    if (gw >= n) return;
    int g = gid[gw];
    const float* hn = h + (size_t)gw * 128;
    for (int c = lane; c < 128; c += 32) atomicAdd(&gsum[g * 128 + c], hn[c]);
    if (lane == 0) atomicAdd(&cnt[g], 1.0f);
}

// wave per graph: mean-pool then MLP
__global__ __launch_bounds__(256) void graph_mlp_k(
    const float* __restrict__ gsum, const float* __restrict__ cnt,
    const float* __restrict__ W1, const float* __restrict__ b1,
    const float* __restrict__ W2, const float* __restrict__ b2,
    float* __restrict__ out, int G) {
    int gw = (int)(((size_t)blockIdx.x * blockDim.x + threadIdx.x) >> 5);
    int lane = threadIdx.x & 31;
    if (gw >= G) return;
    float inv = 1.0f / fmaxf(cnt[gw], 1.0f);
    const float* gs = gsum + (size_t)gw * 128;
    float h0 = b1[2 * lane], h1 = b1[2 * lane + 1];
    for (int k = 0; k < 128; ++k) {
        float v = gs[k] * inv;
        h0 += v * W1[k * 64 + 2 * lane];
        h1 += v * W1[k * 64 + 2 * lane + 1];
    }
    h0 = fmaxf(h0, 0.f);
    h1 = fmaxf(h1, 0.f);
    float p0 = h0 * W2[(2 * lane) * 2 + 0] + h1 * W2[(2 * lane + 1) * 2 + 0];
    float p1 = h0 * W2[(2 * lane) * 2 + 1] + h1 * W2[(2 * lane + 1) * 2 + 1];
    for (int off = 16; off > 0; off >>= 1) {
        p0 += __shfl_down(p0, off, 32);
        p1 += __shfl_down(p1, off, 32);
    }
    if (lane == 0) {
        out[gw * 2 + 0] = p0 + b2[0];
        out[gw * 2 + 1] = p1 + b2[1];
    }
}

// ---------------- host launcher ----------------
static inline unsigned cdiv(size_t a, size_t b) { return (unsigned)((a + b - 1) / b); }

extern "C" void kernel_launch(void* const* d_in, const int* in_sizes, int n_in,
                              void* d_out, int out_size, void* d_ws, size_t ws_size,
                              hipStream_t stream) {
    const float* rf  = (const float*)d_in[0];
    const float* fe  = (const float*)d_in[1];
    const float* cb  = (const float*)d_in[2];
    const float* W1  = (const float*)d_in[3];
    const float* al1 = (const float*)d_in[4];
    const float* ar1 = (const float*)d_in[5];
    const float* b1  = (const float*)d_in[6];
    const float* W2  = (const float*)d_in[7];
    const float* al2 = (const float*)d_in[8];
    const float* ar2 = (const float*)d_in[9];
    const float* b2  = (const float*)d_in[10];
    const float* Wn1 = (const float*)d_in[11];
    const float* bn1 = (const float*)d_in[12];
    const float* Wn2 = (const float*)d_in[13];
    const float* bn2 = (const float*)d_in[14];
    const float* Wg1 = (const float*)d_in[15];
    const float* bg1 = (const float*)d_in[16];
    const float* Wg2 = (const float*)d_in[17];
    const float* bg2 = (const float*)d_in[18];
    const int* src = (const int*)d_in[19];
    const int* dst = (const int*)d_in[20];
    const int* gid = (const int*)d_in[21];
    float* out = (float*)d_out;

    // workspace carve-out (floats)
    float* ws = (float*)d_ws;
    size_t o = 0;
    float* feat1 = ws + o; o += (size_t)NN * H1 * HID;   // [N,4,128]
    float* rst1  = ws + o; o += (size_t)NN * H1 * HID;   // [N,4,128] -> h2 after ELU
    float* feat2 = ws + o; o += (size_t)NN * HID;        // [N,128]
    float* rst2  = ws + o; o += (size_t)NN * HID;        // [N,128] -> h after bias
    float* el1 = ws + o; o += (size_t)NN * H1;
    float* er1 = ws + o; o += (size_t)NN * H1;
    float* m1  = ws + o; o += (size_t)NN * H1;
    float* z1  = ws + o; o += (size_t)NN * H1;
    float* el2 = ws + o; o += (size_t)NN;
    float* er2 = ws + o; o += (size_t)NN;
    float* m2  = ws + o; o += (size_t)NN;
    float* z2  = ws + o; o += (size_t)NN;
    float* gsum = ws + o; o += (size_t)NG * HID;
    float* cnt  = ws + o; o += (size_t)NG;

    const int BT = 256;

    // ---- Layer 1 ----
    {
        dim3 g(cdiv(NN, 128), (H1 * HID) / 128);
        gat_gemm1_wmma<<<g, BT, 0, stream>>>(rf, fe, cb, W1, feat1);
    }
    node_attn<<<cdiv((size_t)NN * H1, BT), BT, 0, stream>>>(feat1, al1, ar1, el1, er1, NN * H1, H1, HID);
    fill_f32<<<cdiv((size_t)NN * H1, BT), BT, 0, stream>>>(m1, -INFINITY, (size_t)NN * H1);
    fill_f32<<<cdiv((size_t)NN * H1, BT), BT, 0, stream>>>(z1, 0.f, (size_t)NN * H1);
    fill_f32<<<cdiv((size_t)NN * H1 * HID, BT), BT, 0, stream>>>(rst1, 0.f, (size_t)NN * H1 * HID);
    edge_max_k   <<<cdiv((size_t)NE * H1, BT), BT, 0, stream>>>(src, dst, el1, er1, m1, NE, H1);
    edge_sumexp_k<<<cdiv((size_t)NE * H1, BT), BT, 0, stream>>>(src, dst, el1, er1, m1, z1, NE, H1);
    edge_aggregate_k<<<cdiv((size_t)NE * H1 * 32, BT), BT, 0, stream>>>(
        src, dst, el1, er1, m1, z1, feat1, rst1, NE, H1, HID);
    bias_act_k<<<cdiv((size_t)NN * H1 * HID, BT), BT, 0, stream>>>(
        rst1, b1, (size_t)NN * H1 * HID, H1 * HID, /*elu=*/1);

    // ---- Layer 2 ----
    {
        dim3 g(cdiv(NN, 128), HID / 128);
        gemm_f32_wmma<<<g, BT, 0, stream>>>(rst1, W2, feat2, NN, HID, H1 * HID);
    }
    node_attn<<<cdiv((size_t)NN, BT), BT, 0, stream>>>(feat2, al2, ar2, el2, er2, NN, 1, HID);
    fill_f32<<<cdiv((size_t)NN, BT), BT, 0, stream>>>(m2, -INFINITY, (size_t)NN);
    fill_f32<<<cdiv((size_t)NN, BT), BT, 0, stream>>>(z2, 0.f, (size_t)NN);
    fill_f32<<<cdiv((size_t)NN * HID, BT), BT, 0, stream>>>(rst2, 0.f, (size_t)NN * HID);
    edge_max_k   <<<cdiv((size_t)NE, BT), BT, 0, stream>>>(src, dst, el2, er2, m2, NE, 1);
    edge_sumexp_k<<<cdiv((size_t)NE, BT), BT, 0, stream>>>(src, dst, el2, er2, m2, z2, NE, 1);
    edge_aggregate_k<<<cdiv((size_t)NE * 32, BT), BT, 0, stream>>>(
        src, dst, el2, er2, m2, z2, feat2, rst2, NE, 1, HID);
    bias_act_k<<<cdiv((size_t)NN * HID, BT), BT, 0, stream>>>(
        rst2, b2, (size_t)NN * HID, HID, /*elu=*/0);

    // ---- heads ----
    fill_f32<<<cdiv((size_t)NG * HID, BT), BT, 0, stream>>>(gsum, 0.f, (size_t)NG * HID);
    fill_f32<<<1, 64, 0, stream>>>(cnt, 0.f, (size_t)NG);
    node_mlp_k<<<cdiv((size_t)NN * 32, BT), BT, 0, stream>>>(rst2, Wn1, bn1, Wn2, bn2, out, NN);
    graph_accum_k<<<cdiv((size_t)NN * 32, BT), BT, 0, stream>>>(rst2, gid, gsum, cnt, NN);
    graph_mlp_k<<<cdiv((size_t)NG * 32, BT), BT, 0, stream>>>(
        gsum, cnt, Wg1, bg1, Wg2, bg2, out + (size_t)NN * 2, NG);
}